// GNN_CT_Classifier_63668595196185
// MI455X (gfx1250) — compile-verified
//
#include <hip/hip_runtime.h>

#define D_INF 128
#define D_H   64
#define N_CLS 2

typedef float v2f __attribute__((ext_vector_type(2)));
typedef float v8f __attribute__((ext_vector_type(8)));

// ---------------- degree / norm ----------------
__global__ void k_deg_init(float* __restrict__ deg, int n) {
  int i = blockIdx.x * blockDim.x + threadIdx.x;
  if (i < n) deg[i] = 1.0f;                      // self-loop
}

__global__ void k_deg_edges(const long long* __restrict__ dst, float* __restrict__ deg, int nE) {
  int e = blockIdx.x * blockDim.x + threadIdx.x;
  if (e < nE) atomicAdd(&deg[(int)dst[e]], 1.0f);
}

__global__ void k_rsqrt(float* __restrict__ deg, int n) {
  int i = blockIdx.x * blockDim.x + threadIdx.x;
  if (i < n) deg[i] = rsqrtf(deg[i]);            // deg >= 1 always
}

__global__ void k_zero4(float4* __restrict__ p, size_t n4) {
  size_t i = (size_t)blockIdx.x * blockDim.x + threadIdx.x;
  if (i < n4) p[i] = make_float4(0.f, 0.f, 0.f, 0.f);
}

// ---------------- dense GEMM via f32 WMMA (16x16x4) ----------------
// One wave32 computes a 16-row x 64-col stripe of H = X(nxK) @ W(KxD_H).
// W is staged block-wide into LDS with gfx1250 async-to-LDS DMA.
template <int K>
__global__ __launch_bounds__(256) void k_gemm_wmma(const float* __restrict__ X,
                                                   const float* __restrict__ W,
                                                   float* __restrict__ H, int n) {
  __shared__ float sW[K * D_H];

  // ---- async stage W -> LDS (16 bytes per thread per step, ASYNCcnt tracked)
  {
    const unsigned nbytes = (unsigned)(K * D_H * sizeof(float));
    const char* gp = (const char*)W;
    for (unsigned ofs = threadIdx.x * 16u; ofs < nbytes; ofs += blockDim.x * 16u) {
      unsigned lds_addr = (unsigned)(size_t)((char*)sW + ofs);          // low 32b = LDS offset
      unsigned long long ga = (unsigned long long)(size_t)(gp + ofs);
      asm volatile("global_load_async_to_lds_b128 %0, %1, off"
                   :: "v"(lds_addr), "v"(ga) : "memory");
    }
    asm volatile("s_wait_asynccnt 0x0" ::: "memory");
    __syncthreads();
  }

  const int lane = threadIdx.x & 31;
  const int wave = blockIdx.x * (blockDim.x >> 5) + (threadIdx.x >> 5);
  const int m0 = wave << 4;
  if (m0 >= n) return;                           // wave-uniform exit

  const int l15 = lane & 15;
  const int hi  = lane >> 4;                     // 0: K-pair {0,1}, 1: {2,3}
  int row = m0 + l15;
  row = row < n ? row : n - 1;                   // clamp: garbage rows never stored
  const float* __restrict__ xr = X + (size_t)row * K;

  v8f acc[4] = {};                               // 16x64 f32 accumulators

#pragma unroll 4
  for (int k = 0; k < K; k += 4) {
    const int ka = k + (hi << 1);
    const v2f a = *reinterpret_cast<const v2f*>(xr + ka);   // 8B-aligned b64 load
#pragma unroll
    for (int nt = 0; nt < 4; ++nt) {
      const float* wp = sW + ka * D_H + (nt << 4) + l15;
      v2f b;
      b.x = wp[0];
      b.y = wp[D_H];                             // +256B -> ds_load_2addr candidate
      acc[nt] = __builtin_amdgcn_wmma_f32_16x16x4_f32(
          false, a, false, b, (short)0, acc[nt], false, false);
    }
  }

#pragma unroll
  for (int nt = 0; nt < 4; ++nt) {
#pragma unroll
    for (int r = 0; r < 8; ++r) {
      const int rr = m0 + r + (hi << 3);         // C/D layout: lanes16-31 -> M+8
      if (rr < n) H[(size_t)rr * D_H + (nt << 4) + l15] = acc[nt][r];
    }
  }
}

// ---------------- sparse aggregation: one wave per edge ----------------
__global__ void k_edge_agg(const long long* __restrict__ src, const long long* __restrict__ dst,
                           const float* __restrict__ dinv, const float* __restrict__ H,
                           float* __restrict__ Agg, int nE) {
  const int lane = threadIdx.x & 31;
  const int e = (int)(((size_t)blockIdx.x * blockDim.x + threadIdx.x) >> 5);
  if (e >= nE) return;
  const int s = (int)src[e];
  const int d = (int)dst[e];
  const float nrm = dinv[s] * dinv[d];
  const float* __restrict__ hs = H + (size_t)s * D_H;
  float* ad = Agg + (size_t)d * D_H;
  atomicAdd(&ad[lane],      hs[lane]      * nrm);
  atomicAdd(&ad[lane + 32], hs[lane + 32] * nrm);
}

// out = (relu?)(Agg + dinv^2 * Hlin + bias)   -- self-loop folded in here
__global__ void k_finalize(const float* __restrict__ Hlin, const float* __restrict__ Agg,
                           const float* __restrict__ dinv, const float* __restrict__ bias,
                           float* __restrict__ Out, int n, int do_relu) {
  size_t idx = (size_t)blockIdx.x * blockDim.x + threadIdx.x;
  if (idx >= (size_t)n * D_H) return;
  const int i = (int)(idx >> 6);
  const int f = (int)(idx & 63);
  const float di = dinv[i];
  float v = Agg[idx] + di * di * Hlin[idx] + bias[f];
  Out[idx] = do_relu ? fmaxf(v, 0.0f) : v;
}

// ---------------- classifier head (64 -> 2) ----------------
__global__ void k_classifier(const float* __restrict__ H, const float* __restrict__ Wfc,
                             const float* __restrict__ bfc, float* __restrict__ out, int n) {
  int idx = blockIdx.x * blockDim.x + threadIdx.x;
  if (idx >= n * N_CLS) return;
  const int i = idx >> 1;
  const int c = idx & 1;
  const float* __restrict__ h = H + (size_t)i * D_H;
  float s = bfc[c];
#pragma unroll
  for (int f = 0; f < D_H; ++f) s += h[f] * Wfc[f * N_CLS + c];
  out[idx] = s;
}

extern "C" void kernel_launch(void* const* d_in, const int* in_sizes, int n_in,
                              void* d_out, int out_size, void* d_ws, size_t ws_size,
                              hipStream_t stream) {
  const float*     x    = (const float*)d_in[0];
  const long long* ei   = (const long long*)d_in[1];
  const float*     W1   = (const float*)d_in[2];
  const float*     b1   = (const float*)d_in[3];
  const float*     W2   = (const float*)d_in[4];
  const float*     b2   = (const float*)d_in[5];
  const float*     Wfc  = (const float*)d_in[6];
  const float*     bfc  = (const float*)d_in[7];
  float*           outp = (float*)d_out;

  const int n  = in_sizes[0] / D_INF;
  const int nE = in_sizes[1] / 2;
  const long long* srcp = ei;
  const long long* dstp = ei + nE;

  // workspace: dinv[n] | bufA[n*64] | bufB[n*64]
  char* ws = (char*)d_ws;
  float* dinv = (float*)ws;
  size_t off = ((size_t)n * sizeof(float) + 255) & ~(size_t)255;
  float* bufA = (float*)(ws + off);
  size_t fsz = (size_t)n * D_H * sizeof(float);
  off = (off + fsz + 255) & ~(size_t)255;
  float* bufB = (float*)(ws + off);

  const size_t nz  = (size_t)n * D_H;
  const int blkN   = (n + 255) / 256;
  const int blkE   = (nE + 255) / 256;
  const int blkZ   = (int)((nz / 4 + 255) / 256);
  const int blkF   = (int)((nz + 255) / 256);
  const int tiles  = (n + 15) / 16;
  const int blkG   = (tiles * 32 + 255) / 256;
  const int blkEA  = (int)(((size_t)nE * 32 + 255) / 256);

  // normalization
  k_deg_init <<<blkN, 256, 0, stream>>>(dinv, n);
  k_deg_edges<<<blkE, 256, 0, stream>>>(dstp, dinv, nE);
  k_rsqrt    <<<blkN, 256, 0, stream>>>(dinv, n);

  // layer 1: A = X @ W1 ; B = scatter(A) ; B = relu(B + dinv^2*A + b1)
  k_gemm_wmma<D_INF><<<blkG, 256, 0, stream>>>(x, W1, bufA, n);
  k_zero4    <<<blkZ, 256, 0, stream>>>((float4*)bufB, nz / 4);
  k_edge_agg <<<blkEA, 256, 0, stream>>>(srcp, dstp, dinv, bufA, bufB, nE);
  k_finalize <<<blkF, 256, 0, stream>>>(bufA, bufB, dinv, b1, bufB, n, 1);

  // layer 2: A = B @ W2 ; B = scatter(A) ; A = B + dinv^2*A + b2
  k_gemm_wmma<D_H><<<blkG, 256, 0, stream>>>(bufB, W2, bufA, n);
  k_zero4    <<<blkZ, 256, 0, stream>>>((float4*)bufB, nz / 4);
  k_edge_agg <<<blkEA, 256, 0, stream>>>(srcp, dstp, dinv, bufA, bufB, nE);
  k_finalize <<<blkF, 256, 0, stream>>>(bufA, bufB, dinv, b2, bufA, n, 0);

  // head
  k_classifier<<<(n * N_CLS + 255) / 256, 256, 0, stream>>>(bufA, Wfc, bfc, outp, n);
}